// PreProcess_31138512896142
// MI455X (gfx1250) — compile-verified
//
#include <hip/hip_runtime.h>
#include <math.h>

#define N 384
#define EPS 1e-8f
#define TILE 32

typedef float v2f __attribute__((ext_vector_type(2)));
typedef float v8f __attribute__((ext_vector_type(8)));

// ---------------------------------------------------------------------------
// Kernel 1: u = x / ||x||_row   (one block per row, 384 threads = 12 waves)
// ---------------------------------------------------------------------------
__global__ __launch_bounds__(N) void rownorm_kernel(const float* __restrict__ x,
                                                    float* __restrict__ u) {
  __shared__ float partial[12];
  const int row = blockIdx.x;
  const int tid = threadIdx.x;
  float v = x[row * N + tid];
  float ss = v * v;
  #pragma unroll
  for (int off = 16; off > 0; off >>= 1) ss += __shfl_down(ss, off, 32);
  if ((tid & 31) == 0) partial[tid >> 5] = ss;
  __syncthreads();
  if (tid == 0) {
    float s = 0.f;
    #pragma unroll
    for (int w = 0; w < 12; ++w) s += partial[w];
    partial[0] = s;
  }
  __syncthreads();
  u[row * N + tid] = v * rsqrtf(partial[0]);
}

// ---------------------------------------------------------------------------
// Kernel 2: G = u * u^T via v_wmma_f32_16x16x4_f32.
// 128 threads = 4 waves; each wave owns a 16x16 tile of a 32x32 block tile.
// LDS stride 33 -> conflict-free column reads of the A/B fragments.
// Fragment layout per ISA 7.12.2: A 16x4 f32 = 2 VGPRs,
//   lane L (L<16): K = 0,1 ; lane L+16: K = 2,3 ; B symmetric (B = U^T).
// D: VGPR m -> (M = m + 8*(lane>>4), N = lane&15).
// ---------------------------------------------------------------------------
__global__ __launch_bounds__(128) void gram_wmma_kernel(const float* __restrict__ u,
                                                        float* __restrict__ G) {
  __shared__ float As[TILE][33];
  __shared__ float Bs[TILE][33];
  const int row0 = blockIdx.y * TILE;
  const int col0 = blockIdx.x * TILE;
  const int tid   = threadIdx.x;
  const int wave  = tid >> 5;
  const int lane  = tid & 31;
  const int wr    = (wave >> 1) * 16;
  const int wc    = (wave & 1) * 16;
  const int lrow  = lane & 15;
  const int lhalf = lane >> 4;

  v8f acc = {};

  for (int kk0 = 0; kk0 < N; kk0 += TILE) {
    if (kk0 + TILE < N) {  // prefetch next K-tile (global_prefetch_b8)
      __builtin_prefetch(u + (row0 + (tid & 31)) * N + kk0 + TILE, 0, 1);
      __builtin_prefetch(u + (col0 + (tid & 31)) * N + kk0 + TILE, 0, 1);
    }
    #pragma unroll
    for (int q = 0; q < 8; ++q) {           // 128 thr * 8 = 1024 = 32x32
      int idx = tid * 8 + q;
      int r = idx >> 5, c = idx & 31;
      As[r][c] = u[(row0 + r) * N + kk0 + c];
      Bs[r][c] = u[(col0 + r) * N + kk0 + c];
    }
    __syncthreads();
    #pragma unroll
    for (int k = 0; k < TILE / 4; ++k) {
      const int kc = 4 * k + 2 * lhalf;
      v2f a, b;
      a.x = As[wr + lrow][kc + 0];
      a.y = As[wr + lrow][kc + 1];
      b.x = Bs[wc + lrow][kc + 0];
      b.y = Bs[wc + lrow][kc + 1];
      acc = __builtin_amdgcn_wmma_f32_16x16x4_f32(
          /*neg_a=*/false, a, /*neg_b=*/false, b,
          /*c_mod=*/(short)0, acc, /*reuse_a=*/false, /*reuse_b=*/false);
    }
    __syncthreads();
  }
  #pragma unroll
  for (int m = 0; m < 8; ++m) {
    int gr = row0 + wr + lhalf * 8 + m;
    int gc = col0 + wc + lrow;
    G[gr * N + gc] = acc[m];
  }
}

// ---------------------------------------------------------------------------
// Kernel 3: c[b,j] from G (analytic cosine-similarity reduction)
// ---------------------------------------------------------------------------
__global__ __launch_bounds__(256) void cosine_kernel(const float* __restrict__ G,
                                                     float* __restrict__ C) {
  int i = blockIdx.x * 256 + threadIdx.x;   // 576 blocks * 256 = N*N exactly
  int b = i / N, j = i - b * N;
  float g  = G[b * N + j];
  float gb = G[b * N];        // G[b,0]
  float gj = G[j * N];        // G[j,0]
  float num = 0.25f * (1.f + gb + g + gj);
  float den = 0.5f * sqrtf(fmaxf(1.f + g, 0.f) * fmaxf(1.f + gb, 0.f));
  C[b * N + j] = num / fmaxf(den, EPS);
}

// ---------------------------------------------------------------------------
// Kernel 4: per batch b, A_b[i,j] = c_b[max(i,j)]  =>  A^-1 = L^T D^-1 L
// (symmetric tridiagonal). Sturm bisection per lane, invert, bitonic sort.
// One block per batch; 512 threads (384 eigen-lanes + pow2 sort padding).
// ---------------------------------------------------------------------------
__global__ __launch_bounds__(512) void eig_kernel(const float* __restrict__ C,
                                                  float* __restrict__ out) {
  __shared__ float alpha[N];
  __shared__ float b2[N];      // b2[i] = beta_{i-1}^2, b2[0] = 0
  __shared__ float buf[512];
  __shared__ float s_lo, s_hi;
  const int b = blockIdx.x;
  const int tid = threadIdx.x;
  const float* a = C + b * N;

  // e_i = 1 / d_i,  d_i = a_i - a_{i+1},  d_{N-1} = a_{N-1}
  if (tid < N) {
    float ai = a[tid];
    float d  = (tid < N - 1) ? (ai - a[tid + 1]) : ai;
    if (fabsf(d) < 1e-30f) d = (d < 0.f) ? -1e-30f : 1e-30f;
    buf[tid] = 1.f / d;
  }
  __syncthreads();
  if (tid < N) {
    float e  = buf[tid];
    float ep = (tid > 0) ? buf[tid - 1] : 0.f;
    alpha[tid] = e + ep;
    b2[tid]    = ep * ep;
  }
  __syncthreads();

  // Gershgorin bounds for T
  float lo_i = INFINITY, hi_i = -INFINITY;
  if (tid < N) {
    float r = sqrtf(b2[tid]) + ((tid < N - 1) ? sqrtf(b2[tid + 1]) : 0.f);
    lo_i = alpha[tid] - r;
    hi_i = alpha[tid] + r;
  }
  buf[tid] = lo_i;
  __syncthreads();
  for (int s = 256; s > 0; s >>= 1) {
    if (tid < s) buf[tid] = fminf(buf[tid], buf[tid + s]);
    __syncthreads();
  }
  if (tid == 0) s_lo = buf[0] - 1e-3f * fabsf(buf[0]) - 1e-6f;
  __syncthreads();
  buf[tid] = hi_i;
  __syncthreads();
  for (int s = 256; s > 0; s >>= 1) {
    if (tid < s) buf[tid] = fmaxf(buf[tid], buf[tid + s]);
    __syncthreads();
  }
  if (tid == 0) s_hi = buf[0] + 1e-3f * fabsf(buf[0]) + 1e-6f;
  __syncthreads();

  // Bisection: lane tid finds the tid-th smallest eigenvalue of T
  float val = INFINITY;
  if (tid < N) {
    float lo = s_lo, hi = s_hi;
    for (int it = 0; it < 50; ++it) {
      float mid = 0.5f * (lo + hi);
      float q = alpha[0] - mid;
      int cnt = (q < 0.f) ? 1 : 0;
      for (int i2 = 1; i2 < N; ++i2) {         // LDS-broadcast friendly scan
        float dq = q;
        if (fabsf(dq) < 1e-30f) dq = (dq < 0.f) ? -1e-30f : 1e-30f;
        q = (alpha[i2] - mid) - b2[i2] / dq;
        cnt += (q < 0.f) ? 1 : 0;
      }
      if (cnt <= tid) lo = mid; else hi = mid;
    }
    float lamT = 0.5f * (lo + hi);
    if (fabsf(lamT) < 1e-30f) lamT = (lamT < 0.f) ? -1e-30f : 1e-30f;
    val = 1.f / lamT;                           // eig(A) = 1 / eig(T)
  }
  __syncthreads();
  buf[tid] = val;                               // pad lanes carry +INF
  __syncthreads();

  // bitonic sort, 512 elements, ascending (eigvalsh order)
  for (int ks = 2; ks <= 512; ks <<= 1) {
    for (int j = ks >> 1; j > 0; j >>= 1) {
      int ixj = tid ^ j;
      if (ixj > tid) {
        bool asc = ((tid & ks) == 0);
        float x0 = buf[tid], x1 = buf[ixj];
        if ((x0 > x1) == asc) { buf[tid] = x1; buf[ixj] = x0; }
      }
      __syncthreads();
    }
  }
  if (tid < N) out[b * N + tid] = buf[tid];
}

// ---------------------------------------------------------------------------
extern "C" void kernel_launch(void* const* d_in, const int* in_sizes, int n_in,
                              void* d_out, int out_size, void* d_ws, size_t ws_size,
                              hipStream_t stream) {
  const float* x = (const float*)d_in[0];
  float* out = (float*)d_out;
  float* u = (float*)d_ws;          // N*N
  float* G = u + N * N;             // N*N
  float* C = G + N * N;             // N*N   (total 1.73 MB workspace)

  rownorm_kernel<<<N, N, 0, stream>>>(x, u);
  dim3 g2(N / TILE, N / TILE);
  gram_wmma_kernel<<<g2, 128, 0, stream>>>(u, G);
  cosine_kernel<<<(N * N) / 256, 256, 0, stream>>>(G, C);
  eig_kernel<<<N, 512, 0, stream>>>(C, out);
}